// Processor_58308476010998
// MI455X (gfx1250) — compile-verified
//
#include <hip/hip_runtime.h>
#include <stdint.h>

typedef __attribute__((ext_vector_type(2))) float v2f;
typedef __attribute__((ext_vector_type(4))) float v4f;
typedef __attribute__((ext_vector_type(8))) float v8f;

#define TILE 128
#define KB   32
#define LDA  36    // A-tile row stride (floats): 144B rows -> 16B aligned, conflict-free b64 frags
#define LDB  144   // K-major tile row stride (floats): 576B rows, 2*LDB%64banks=32 -> conflict-free

// gfx1250 async global->LDS copy (ASYNCcnt); LDS offset = low 32 bits of generic addr.
__device__ __forceinline__ void async_cp_b128(unsigned lds_off, const float* g) {
  asm volatile("global_load_async_to_lds_b128 %0, %1, off"
               :: "v"(lds_off), "v"((unsigned long long)g) : "memory");
}
__device__ __forceinline__ void wait_async0() {
  asm volatile("s_wait_asynccnt 0x0" ::: "memory");
}
__device__ __forceinline__ unsigned lds_off32(const void* p) {
  return (unsigned)(uintptr_t)p;
}

// ---------------------------------------------------------------------------
// Batched f32 GEMM: C[b] = X[b] * Y[b], n x n, V_WMMA_F32_16X16X4_F32.
// 256 threads = 8 waves; macro tile 128x128; wave tile 32(M) x 64(N).
// Both tiles staged with double-buffered GLOBAL_LOAD_ASYNC_TO_LDS_B128:
// one s_wait_asynccnt + one barrier per K-step, copies overlap the WMMA burst.
// ---------------------------------------------------------------------------
__global__ __launch_bounds__(256)
void gemm_sq_kernel(const float* __restrict__ X, const float* __restrict__ Y,
                    float* __restrict__ C, int n,
                    long long sx, long long sy, long long sc) {
  __shared__ float sA[2][TILE * LDA];  // [M=128][K=32] row-major
  __shared__ float sB[2][KB * LDB];    // [K=32][N=128] row-major (no transpose)

  const int bz = blockIdx.z;
  const float* Xb = X + (long long)bz * sx;
  const float* Yb = Y + (long long)bz * sy;
  float*       Cb = C + (long long)bz * sc;

  const int mBase = blockIdx.x * TILE;
  const int nBase = blockIdx.y * TILE;
  const int t    = threadIdx.x;
  const int wave = t >> 5, lane = t & 31, l16 = lane & 15, lh = lane >> 4;
  const int wm = wave & 3, wn = wave >> 2;

  // Per-thread staging slots (computed once; advanced by constant offsets only).
  const int xrow = t >> 3, xc4 = t & 7;    // A tile: 8 float4 per 32-float row
  const int yrow = t >> 5, yc4 = t & 31;   // B tile: 32 float4 per 128-float row
  const float* gx = Xb + (long long)(mBase + xrow) * n + xc4 * 4;
  const float* gy = Yb + (long long)yrow * n + nBase + yc4 * 4;
  const unsigned lA0 = lds_off32(&sA[0][xrow * LDA + xc4 * 4]);
  const unsigned lB0 = lds_off32(&sB[0][yrow * LDB + yc4 * 4]);
  const unsigned strideA = sizeof(float) * TILE * LDA;
  const unsigned strideB = sizeof(float) * KB * LDB;

  auto stage = [&](int buf, int ko) {
    const float* gxk = gx + ko;                  // advance along K (cols of X)
    const float* gyk = gy + (long long)ko * n;   // advance along K (rows of Y)
    const unsigned la = lA0 + (unsigned)buf * strideA;
    const unsigned lb = lB0 + (unsigned)buf * strideB;
#pragma unroll
    for (int r = 0; r < 4; ++r) {
      async_cp_b128(la + r * (32 * LDA * 4), gxk + (long long)r * 32 * n);
      async_cp_b128(lb + r * (8 * LDB * 4),  gyk + (long long)r * 8 * n);
    }
  };

  v8f acc[2][4];
#pragma unroll
  for (int i = 0; i < 2; ++i)
#pragma unroll
    for (int j = 0; j < 4; ++j) acc[i][j] = {};

  stage(0, 0);
  for (int ko = 0; ko < n; ko += KB) {
    const int cur = (ko >> 5) & 1;
    wait_async0();          // this wave's copies into `cur` done
    __syncthreads();        // all waves' copies done + prev buffer free
    if (ko + KB < n) stage(cur ^ 1, ko + KB);

    const float* pA = &sA[cur][0];
    const float* pB = &sB[cur][0];
#pragma unroll
    for (int kk = 0; kk < KB; kk += 4) {
      v2f af[2], bf[4];
#pragma unroll
      for (int i = 0; i < 2; ++i)
        af[i] = *(const v2f*)(pA + (wm * 32 + i * 16 + l16) * LDA + kk + 2 * lh);
#pragma unroll
      for (int j = 0; j < 4; ++j) {
        const float* p = pB + (kk + 2 * lh) * LDB + wn * 64 + j * 16 + l16;
        v2f tmp; tmp.x = p[0]; tmp.y = p[LDB];   // -> ds_load_2addr_b32
        bf[j] = tmp;
      }
#pragma unroll
      for (int i = 0; i < 2; ++i)
#pragma unroll
        for (int j = 0; j < 4; ++j)
          acc[i][j] = __builtin_amdgcn_wmma_f32_16x16x4_f32(
              false, af[i], false, bf[j], (short)0, acc[i][j], false, false);
    }
  }

#pragma unroll
  for (int i = 0; i < 2; ++i)
#pragma unroll
    for (int j = 0; j < 4; ++j) {
      int col = nBase + wn * 64 + j * 16 + l16;
#pragma unroll
      for (int v = 0; v < 8; ++v) {
        int row = mBase + wm * 32 + i * 16 + lh * 8 + v;
        Cb[(long long)row * n + col] = acc[i][j][v];
      }
    }
}

// ---------------------------------------------------------------------------
// Masked aggregation: agg[b,i,d] (+)= sum_j (Wp[b,j,i]*(Apow[j,i]==1)) * x[b,j,d]
// Both tiles kept K-major (no transpose): A-fragment reads sW[j][i] directly.
// x tile streamed with async copies; masked tile through VALU + b128 DS store.
// ---------------------------------------------------------------------------
__global__ __launch_bounds__(256)
void agg_kernel(const float* __restrict__ Wp, const float* __restrict__ Apow,
                const float* __restrict__ x, float* __restrict__ agg,
                int n, int d, long long sWp, long long sAp, int accumulate) {
  __shared__ float sW[KB * LDB];   // masked tile [j=0..31][i=0..127]
  __shared__ float sX[KB * LDB];   // x tile      [j=0..31][d=0..127]

  const int bz = blockIdx.z;
  const float* Wb = Wp + (long long)bz * sWp;
  const float* Ab = Apow + (long long)bz * sAp;
  const float* xb = x + (long long)bz * n * d;
  float*       gb = agg + (long long)bz * n * d;

  const int mBase = blockIdx.x * TILE;   // i block
  const int t    = threadIdx.x;
  const int wave = t >> 5, lane = t & 31, l16 = lane & 15, lh = lane >> 4;
  const int wm = wave & 3, wn = wave >> 2;

  const int row = t >> 5, c4 = t & 31;   // staging map for 32x128 tiles
  const float* gxp = xb + (long long)row * d + c4 * 4;
  const unsigned lX0 = lds_off32(&sX[row * LDB + c4 * 4]);

  v8f acc[2][4];
#pragma unroll
  for (int i = 0; i < 2; ++i)
#pragma unroll
    for (int j = 0; j < 4; ++j) acc[i][j] = {};

  for (int ko = 0; ko < n; ko += KB) {
    // async-stage x tile rows ko..ko+31
#pragma unroll
    for (int r = 0; r < 4; ++r)
      async_cp_b128(lX0 + r * (8 * LDB * 4), gxp + (long long)(ko + 8 * r) * d);
    // masked Wp tile (mask applied in VALU, aligned b128 LDS store)
#pragma unroll
    for (int r = 0; r < 4; ++r) {
      long long off = (long long)(ko + row + 8 * r) * n + mBase + c4 * 4;
      v4f wv = *(const v4f*)(Wb + off);
      v4f av = *(const v4f*)(Ab + off);
      v4f m;
#pragma unroll
      for (int e = 0; e < 4; ++e) m[e] = (av[e] == 1.0f) ? wv[e] : 0.0f;
      *(v4f*)(&sW[(row + 8 * r) * LDB + c4 * 4]) = m;
    }
    wait_async0();
    __syncthreads();

#pragma unroll
    for (int kk = 0; kk < KB; kk += 4) {
      v2f af[2], bf[4];
#pragma unroll
      for (int i = 0; i < 2; ++i) {
        const float* p = sW + (kk + 2 * lh) * LDB + wm * 32 + i * 16 + l16;
        v2f tmp; tmp.x = p[0]; tmp.y = p[LDB];
        af[i] = tmp;
      }
#pragma unroll
      for (int j = 0; j < 4; ++j) {
        const float* p = sX + (kk + 2 * lh) * LDB + wn * 64 + j * 16 + l16;
        v2f tmp; tmp.x = p[0]; tmp.y = p[LDB];
        bf[j] = tmp;
      }
#pragma unroll
      for (int i = 0; i < 2; ++i)
#pragma unroll
        for (int j = 0; j < 4; ++j)
          acc[i][j] = __builtin_amdgcn_wmma_f32_16x16x4_f32(
              false, af[i], false, bf[j], (short)0, acc[i][j], false, false);
    }
    __syncthreads();
  }

#pragma unroll
  for (int i = 0; i < 2; ++i)
#pragma unroll
    for (int j = 0; j < 4; ++j) {
      int col = wn * 64 + j * 16 + l16;   // d index
#pragma unroll
      for (int v = 0; v < 8; ++v) {
        int r2 = mBase + wm * 32 + i * 16 + lh * 8 + v;
        long long idx = (long long)r2 * d + col;
        float val = acc[i][j][v];
        if (accumulate) val += gb[idx];
        gb[idx] = val;
      }
    }
}

// ---------------------------------------------------------------------------
// Final: h = agg @ W^T + b + x; LayerNorm with unbiased std (ddof=1), eps on std.
// ---------------------------------------------------------------------------
__global__ __launch_bounds__(128)
void final_kernel(const float* __restrict__ agg, const float* __restrict__ x,
                  const float* __restrict__ W, const float* __restrict__ bias,
                  const float* __restrict__ a2, const float* __restrict__ b2,
                  float* __restrict__ y, int d) {
  __shared__ float sAgg[128];
  __shared__ float sRed[4];

  const long long row = blockIdx.x;
  const int t = threadIdx.x;
  sAgg[t] = agg[row * d + t];
  __syncthreads();

  float accv = bias[t];
#pragma unroll 8
  for (int k = 0; k < 128; ++k) accv = fmaf(sAgg[k], W[t * d + k], accv);
  float h = accv + x[row * d + t];

  float s = h;
#pragma unroll
  for (int off = 16; off > 0; off >>= 1) s += __shfl_down(s, off, 32);
  const int wv = t >> 5, ln = t & 31;
  if (ln == 0) sRed[wv] = s;
  __syncthreads();
  float mean = (sRed[0] + sRed[1] + sRed[2] + sRed[3]) * (1.0f / 128.0f);

  float dv = h - mean;
  float q = dv * dv;
#pragma unroll
  for (int off = 16; off > 0; off >>= 1) q += __shfl_down(q, off, 32);
  __syncthreads();
  if (ln == 0) sRed[wv] = q;
  __syncthreads();
  float var = (sRed[0] + sRed[1] + sRed[2] + sRed[3]) * (1.0f / 127.0f);
  float stdv = sqrtf(var);

  y[row * d + t] = a2[t] * dv / (stdv + 1e-6f) + b2[t];
}

extern "C" void kernel_launch(void* const* d_in, const int* in_sizes, int n_in,
                              void* d_out, int out_size, void* d_ws, size_t ws_size,
                              hipStream_t stream) {
  (void)in_sizes; (void)n_in; (void)out_size; (void)ws_size;
  const float* x    = (const float*)d_in[0];   // [B,N,D]
  const float* adjw = (const float*)d_in[1];   // [B,N,N]
  const float* adj  = (const float*)d_in[2];   // [N,N]
  const float* W    = (const float*)d_in[3];   // [D,D]
  const float* bias = (const float*)d_in[4];   // [D]
  const float* a2   = (const float*)d_in[5];   // [D]
  const float* b2   = (const float*)d_in[6];   // [D]
  float* y = (float*)d_out;

  const int B = 32, N = 1024, D = 128;
  const long long NN = (long long)N * N;

  float* A1  = (float*)d_ws;                 //  4 MB
  float* A2  = A1 + NN;                      //  4 MB
  float* agg = A2 + NN;                      // 16 MB
  float* Wp1 = agg + (long long)B * N * D;   // 128 MB
  float* Wp2 = Wp1 + (long long)B * NN;      // 128 MB

  dim3 blk(256);
  dim3 g1(N / TILE, N / TILE, 1);
  dim3 gB(N / TILE, N / TILE, B);
  dim3 gA(N / TILE, 1, B);

  // Adjacency powers (exact integer path counts in f32)
  gemm_sq_kernel<<<g1, blk, 0, stream>>>(adj, adj, A1, N, 0, 0, 0);
  gemm_sq_kernel<<<g1, blk, 0, stream>>>(A1, A1, A2, N, 0, 0, 0);
  // Batched weight squarings (dominant cost: 137 GFLOP f32)
  gemm_sq_kernel<<<gB, blk, 0, stream>>>(adjw, adjw, Wp1, N, NN, NN, NN);
  gemm_sq_kernel<<<gB, blk, 0, stream>>>(Wp1, Wp1, Wp2, N, NN, NN, NN);
  // Masked aggregations (hop 0 uses raw adj; batch stride 0 broadcasts it)
  agg_kernel<<<gA, blk, 0, stream>>>(adjw, adj, x, agg, N, D, NN, 0, 0);
  agg_kernel<<<gA, blk, 0, stream>>>(Wp1, A1, x, agg, N, D, NN, 0, 1);
  agg_kernel<<<gA, blk, 0, stream>>>(Wp2, A2, x, agg, N, D, NN, 0, 1);
  // Linear + residual + LayerNorm
  final_kernel<<<dim3(B * N), dim3(128), 0, stream>>>(agg, x, W, bias, a2, b2, y, D);
}